// CriticMpnn_12446815223932
// MI455X (gfx1250) — compile-verified
//
#include <hip/hip_runtime.h>

typedef __attribute__((ext_vector_type(2))) float v2f;
typedef __attribute__((ext_vector_type(8))) float v8f;

#define LEAK 0.01f

// lrelu(x) == max(x, LEAK*x) exactly, since 0 < LEAK < 1  (2 VALU ops)
__device__ __forceinline__ float lrelu(float x) { return fmaxf(x, LEAK * x); }

// Order-preserving float -> uint key (smaller float <=> smaller uint), branch-free
__device__ __forceinline__ unsigned fkey(float f) {
    unsigned u = __float_as_uint(f);
    return u ^ ((unsigned)((int)u >> 31) | 0x80000000u);
}
__device__ __forceinline__ float keyf(unsigned k) {
    return __uint_as_float(k ^ (~(unsigned)((int)k >> 31) | 0x80000000u));
}
__device__ __forceinline__ bool finitef(float f) {
    return (__float_as_uint(f) & 0x7f800000u) != 0x7f800000u;
}

// x0 = [float(nf0), float(nf1), action]
__global__ void build_x0(const int* __restrict__ nf, const float* __restrict__ act,
                         float* __restrict__ x0, int total) {
    int i = blockIdx.x * blockDim.x + threadIdx.x;
    if (i < total) {
        x0[i * 3 + 0] = (float)nf[i * 2 + 0];
        x0[i * 3 + 1] = (float)nf[i * 2 + 1];
        x0[i * 3 + 2] = act[i];
    }
}

// One wave processes TILES tiles of 16 edges. Per-edge MLP via
// V_WMMA_F32_16X16X4_F32 (fp32 fidelity), exact scatter-min at dst.
// GUARD=false: no bounds checks (caller guarantees all edges in range).
template <int FIN, int TILES, bool GUARD>
__global__ __launch_bounds__(32) void edge_layer(
    const float* __restrict__ xin_all,     // [B,N,FIN]
    const long long* __restrict__ ei,      // [B,2,E]  (row0=src, row1=dst)
    const float* __restrict__ ea_all,      // [B,E]
    const float* __restrict__ W1, const float* __restrict__ b1,   // [2*FIN+1,8],[8]
    const float* __restrict__ W2, const float* __restrict__ b2,   // [8,8],[8]
    unsigned* __restrict__ agg_all,        // [B,N,8] key-encoded running min
    int N, int E, long long ebase)
{
    constexpr int K1  = 2 * FIN + 1;
    constexpr int K1P = (K1 + 3) & ~3;
    constexpr int NCH = K1P / 4;

    const int b    = blockIdx.y;
    const int lane = threadIdx.x;     // 0..31
    const int i    = lane & 15;       // tile-local edge / A-matrix row
    const int n    = lane & 15;       // output column (N dim)
    const int koff = (lane >> 4) * 2; // K sub-offset per lane half (A and B)

    const float*     xin  = xin_all + (size_t)b * N * FIN;
    const long long* srcp = ei + (size_t)b * 2 * E;
    const long long* dstp = srcp + E;
    const float*     ea   = ea_all + (size_t)b * E;
    unsigned*        agg  = agg_all + (size_t)b * N * 8;   // wave-uniform base

    __shared__ alignas(16) float m[16][K1P];   // 16B-aligned rows (K1P multiple of 4)
    __shared__ alignas(16) float hbuf[16][8];
    __shared__ alignas(16) int   dbuf[16];

    // ---- loop-invariant weight fragments (B operands) + biases ----
    v2f wb1[NCH];
    #pragma unroll
    for (int c = 0; c < NCH; ++c) {
        int kb = c * 4 + koff;
        wb1[c].x = (kb     < K1 && n < 8) ? W1[kb * 8 + n]       : 0.f;
        wb1[c].y = (kb + 1 < K1 && n < 8) ? W1[(kb + 1) * 8 + n] : 0.f;
    }
    v2f wb2[2];
    #pragma unroll
    for (int c = 0; c < 2; ++c) {
        int kb = c * 4 + koff;
        wb2[c].x = (n < 8) ? W2[kb * 8 + n]       : 0.f;
        wb2[c].y = (n < 8) ? W2[(kb + 1) * 8 + n] : 0.f;
    }
    const float bias1 = (n < 8) ? b1[n] : 0.f;
    const float bias2 = (n < 8) ? b2[n] : 0.f;
    const int   mrow  = (lane < 16) ? 0 : 8;

    const long long base = ebase + (long long)blockIdx.x * (16 * TILES);

    for (int t = 0; t < TILES; ++t) {
        const long long e0 = base + (long long)t * 16;   // wave-uniform
        if (GUARD && e0 >= E) break;                     // uniform exit: EXEC stays full

        // ---- gather: build 16 messages [x_dst | x_src | ea | 0-pad] ----
        long long e = e0 + i;
        if (lane < 16) {
            bool ok = !GUARD || (e < E);
            if (ok) {
                int d = (int)dstp[e];
                dbuf[i] = d;
                if constexpr (FIN == 8) {
                    const float4* xp = (const float4*)(xin + (size_t)d * 8);
                    *(float4*)&m[i][0] = xp[0];
                    *(float4*)&m[i][4] = xp[1];
                } else {
                    #pragma unroll
                    for (int f = 0; f < FIN; ++f) m[i][f] = xin[(size_t)d * FIN + f];
                }
                m[i][2 * FIN] = ea[e];
            } else {
                dbuf[i] = -1;
                #pragma unroll
                for (int f = 0; f < FIN; ++f) m[i][f] = 0.f;
                m[i][2 * FIN] = 0.f;
            }
            #pragma unroll
            for (int k = K1; k < K1P; ++k) m[i][k] = 0.f;
        } else {
            bool ok = !GUARD || (e < E);
            if (ok) {
                int s = (int)srcp[e];
                if constexpr (FIN == 8) {
                    const float4* xp = (const float4*)(xin + (size_t)s * 8);
                    *(float4*)&m[i][8]  = xp[0];
                    *(float4*)&m[i][12] = xp[1];
                } else {
                    #pragma unroll
                    for (int f = 0; f < FIN; ++f) m[i][FIN + f] = xin[(size_t)s * FIN + f];
                }
            } else {
                #pragma unroll
                for (int f = 0; f < FIN; ++f) m[i][FIN + f] = 0.f;
            }
        }
        __syncthreads();   // single-wave workgroup: cheap

        // destination ids for this lane's 8 output rows: two ds_load_b128
        int4 d03 = *(const int4*)&dbuf[mrow];
        int4 d47 = *(const int4*)&dbuf[mrow + 4];
        int dvec[8] = { d03.x, d03.y, d03.z, d03.w, d47.x, d47.y, d47.z, d47.w };

        // ---- GEMM1: [16 x K1P] @ [K1P x 16], bias carried in C ----
        v8f acc;
        #pragma unroll
        for (int r = 0; r < 8; ++r) acc[r] = bias1;
        #pragma unroll
        for (int c = 0; c < NCH; ++c) {
            int kb = c * 4 + koff;
            v2f a; a.x = m[i][kb]; a.y = m[i][kb + 1];
            acc = __builtin_amdgcn_wmma_f32_16x16x4_f32(false, a, false, wb1[c],
                                                        (short)0, acc, false, false);
        }

        // ---- LeakyReLU, re-stage hidden [16 x 8] as GEMM2 A operand ----
        if (n < 8) {
            #pragma unroll
            for (int r = 0; r < 8; ++r) hbuf[mrow + r][n] = lrelu(acc[r]);
        }
        __syncthreads();

        // ---- GEMM2: [16 x 8] @ [8 x 16], bias carried in C ----
        v8f acc2;
        #pragma unroll
        for (int r = 0; r < 8; ++r) acc2[r] = bias2;
        #pragma unroll
        for (int c = 0; c < 2; ++c) {
            int kb = c * 4 + koff;
            v2f a; a.x = hbuf[i][kb]; a.y = hbuf[i][kb + 1];
            acc2 = __builtin_amdgcn_wmma_f32_16x16x4_f32(false, a, false, wb2[c],
                                                         (short)0, acc2, false, false);
        }

        // ---- scatter-min at dst: uniform 64-bit base + 32-bit lane offset (GVS) ----
        if (n < 8) {
            #pragma unroll
            for (int r = 0; r < 8; ++r) {
                int d = dvec[r];
                unsigned off = ((unsigned)d << 3) | (unsigned)n;   // d*8+n < 2^31
                if (!GUARD || d >= 0)
                    atomicMin(agg + off, fkey(acc2[r]));
            }
        }
        __syncthreads();   // protect LDS reuse across tiles
    }
}

// Decode min-keys, empty->0, LeakyReLU, reset agg. 4 elements/thread (b128 path).
__global__ void finalize_layer(unsigned* __restrict__ agg, float* __restrict__ xout, int total4) {
    int idx = blockIdx.x * blockDim.x + threadIdx.x;
    if (idx < total4) {
        uint4 k = ((const uint4*)agg)[idx];
        float4 o;
        float f0 = keyf(k.x); o.x = finitef(f0) ? lrelu(f0) : 0.f;
        float f1 = keyf(k.y); o.y = finitef(f1) ? lrelu(f1) : 0.f;
        float f2 = keyf(k.z); o.z = finitef(f2) ? lrelu(f2) : 0.f;
        float f3 = keyf(k.w); o.w = finitef(f3) ? lrelu(f3) : 0.f;
        ((float4*)xout)[idx] = o;
        ((uint4*)agg)[idx] = make_uint4(0xffffffffu, 0xffffffffu, 0xffffffffu, 0xffffffffu);
    }
}

// Deterministic two-stage sum pool of [x0(3) | h(8)].
__global__ void pool_partial(const float* __restrict__ x0, const float* __restrict__ h,
                             float* __restrict__ partial, int N, int nblk) {
    int b = blockIdx.y, blk = blockIdx.x, tid = threadIdx.x;
    float loc[11];
    #pragma unroll
    for (int j = 0; j < 11; ++j) loc[j] = 0.f;
    for (int nd = blk * blockDim.x + tid; nd < N; nd += nblk * blockDim.x) {
        const float* xp = x0 + ((size_t)b * N + nd) * 3;
        const float4* hp = (const float4*)(h + ((size_t)b * N + nd) * 8);
        float4 h0 = hp[0], h1 = hp[1];
        loc[0] += xp[0]; loc[1] += xp[1]; loc[2] += xp[2];
        loc[3] += h0.x; loc[4] += h0.y; loc[5] += h0.z; loc[6] += h0.w;
        loc[7] += h1.x; loc[8] += h1.y; loc[9] += h1.z; loc[10] += h1.w;
    }
    __shared__ float red[256];
    for (int j = 0; j < 11; ++j) {
        red[tid] = loc[j];
        __syncthreads();
        for (int s = 128; s > 0; s >>= 1) {
            if (tid < s) red[tid] += red[tid + s];
            __syncthreads();
        }
        if (tid == 0) partial[((size_t)b * nblk + blk) * 11 + j] = red[0];
        __syncthreads();
    }
}

__global__ void final_out(const float* __restrict__ partial, const float* __restrict__ linW,
                          const float* __restrict__ linb, float* __restrict__ out,
                          int B, int nblk) {
    __shared__ float pool[8][11];
    int tid = threadIdx.x;          // 64 threads: b = tid/16, j = tid%16
    int b = tid >> 4, j = tid & 15;
    if (b < B && j < 11) {
        float s = 0.f;
        for (int k = 0; k < nblk; ++k) s += partial[((size_t)b * nblk + k) * 11 + j];
        pool[b][j] = s;
    }
    __syncthreads();
    if (tid < B) {
        float s = linb[0];
        for (int j2 = 0; j2 < 11; ++j2) s += pool[tid][j2] * linW[j2];
        out[tid] = s;
    }
}

extern "C" void kernel_launch(void* const* d_in, const int* in_sizes, int n_in,
                              void* d_out, int out_size, void* d_ws, size_t ws_size,
                              hipStream_t stream) {
    const int*       nf   = (const int*)d_in[0];
    const float*     act  = (const float*)d_in[1];
    const long long* ei   = (const long long*)d_in[2];
    const float*     ef   = (const float*)d_in[3];
    const float* W[12];
    for (int k = 0; k < 12; ++k) W[k] = (const float*)d_in[4 + k];
    const float *linW = (const float*)d_in[16], *linb = (const float*)d_in[17];
    float* out = (float*)d_out;

    const int B = out_size;              // output is [B,1]
    const int N = in_sizes[1] / B;       // actions: [B,N]
    const int E = in_sizes[3] / B;       // edge_features: [B,E,1]

    char* ws = (char*)d_ws;
    size_t off = 0;
    float*    x0  = (float*)(ws + off);    off += (size_t)B * N * 3 * sizeof(float);
    float*    h1  = (float*)(ws + off);    off += (size_t)B * N * 8 * sizeof(float);
    float*    h2  = (float*)(ws + off);    off += (size_t)B * N * 8 * sizeof(float);
    unsigned* agg = (unsigned*)(ws + off); off += (size_t)B * N * 8 * sizeof(unsigned);
    float* partial = (float*)(ws + off);
    const int nblk = (N + 255) / 256;

    build_x0<<<(B * N + 255) / 256, 256, 0, stream>>>(nf, act, x0, B * N);
    hipMemsetAsync(agg, 0xFF, (size_t)B * N * 8 * sizeof(unsigned), stream);

    constexpr int TILES = 8;
    const int epb    = 16 * TILES;        // edges per block
    const int nfull  = E / epb;           // fully in-range blocks
    const long long tail_base = (long long)nfull * epb;
    const bool has_tail = (tail_base < E);
    const int tot4 = B * N * 8 / 4;       // N*8 divisible by 4

    auto run_layer = [&](int fin, const float* xin, int wbase, float* hout) {
        if (nfull > 0) {
            dim3 eg((unsigned)nfull, (unsigned)B);
            if (fin == 3)
                edge_layer<3, TILES, false><<<eg, 32, 0, stream>>>(
                    xin, ei, ef, W[wbase], W[wbase+1], W[wbase+2], W[wbase+3], agg, N, E, 0);
            else
                edge_layer<8, TILES, false><<<eg, 32, 0, stream>>>(
                    xin, ei, ef, W[wbase], W[wbase+1], W[wbase+2], W[wbase+3], agg, N, E, 0);
        }
        if (has_tail) {
            dim3 eg(1u, (unsigned)B);
            if (fin == 3)
                edge_layer<3, TILES, true><<<eg, 32, 0, stream>>>(
                    xin, ei, ef, W[wbase], W[wbase+1], W[wbase+2], W[wbase+3], agg, N, E, tail_base);
            else
                edge_layer<8, TILES, true><<<eg, 32, 0, stream>>>(
                    xin, ei, ef, W[wbase], W[wbase+1], W[wbase+2], W[wbase+3], agg, N, E, tail_base);
        }
        finalize_layer<<<(tot4 + 255) / 256, 256, 0, stream>>>(agg, hout, tot4);
    };

    run_layer(3, x0, 0, h1);   // c1
    run_layer(8, h1, 4, h2);   // c2
    run_layer(8, h2, 8, h1);   // c3

    pool_partial<<<dim3((unsigned)nblk, (unsigned)B), 256, 0, stream>>>(x0, h1, partial, N, nblk);
    final_out<<<1, 64, 0, stream>>>(partial, linW, linb, out, B, nblk);
}